// GraphChannelMixerPyG_49821620634208
// MI455X (gfx1250) — compile-verified
//
#include <hip/hip_runtime.h>
#include <math.h>

#define B_   32
#define N_   19
#define T_   512
#define D_   64
#define KEIG 16
#define NC0  80        // D + K_EIG
#define G_   (B_ * T_)

#define SA 21          // LDS row stride for 19x19 matrices (conflict-free, gcd(21,64)=1)
#define SX 84          // LDS row stride for 19x80 buffers
#define SHS 68         // LDS row stride for 19x64 buffer

typedef float v2f __attribute__((ext_vector_type(2)));
typedef float v8f __attribute__((ext_vector_type(8)));

__device__ __forceinline__ float softplus_f(float z) { return log1pf(expf(z)); }
__device__ __forceinline__ float gelu_f(float x) {
    return 0.5f * x * (1.0f + erff(x * 0.70710678118654752f));
}

// -----------------------------------------------------------------------------
// Out(19 x NC) = Tn^T (19x19) @ In(19 x NC)   via V_WMMA_F32_16X16X4_F32
// A(m,k) = Tn[k*SA + m]  (transpose read), B(k,n) = In[k*in_rs + n]
// -----------------------------------------------------------------------------
template <int NC>
__device__ __forceinline__ void matmulT(int lane, const float* __restrict__ Tn,
                                        const float* __restrict__ In, int in_rs,
                                        float* __restrict__ Out /* stride SX */) {
    const int mloc = lane & 15;
    const int kk   = (lane >> 4) << 1;   // 0 for lanes 0-15, 2 for lanes 16-31
    for (int m0 = 0; m0 < N_; m0 += 16) {
        for (int n0 = 0; n0 < NC; n0 += 16) {
            v8f acc = {0.f, 0.f, 0.f, 0.f, 0.f, 0.f, 0.f, 0.f};
            const int n = n0 + mloc;           // NC multiple of 16 -> always in range
            for (int k0 = 0; k0 < N_; k0 += 4) {
                const int m   = m0 + mloc;
                const int ka  = k0 + kk;
                const int kb  = ka + 1;
                const int mcl = (m < N_) ? m : (N_ - 1);
                const int kac = (ka < N_) ? ka : (N_ - 1);
                const int kbc = (kb < N_) ? kb : (N_ - 1);
                v2f a, b;
                float ax = Tn[kac * SA + mcl];
                float ay = Tn[kbc * SA + mcl];
                a.x = (m < N_ && ka < N_) ? ax : 0.f;
                a.y = (m < N_ && kb < N_) ? ay : 0.f;
                float bx = In[kac * in_rs + n];
                float by = In[kbc * in_rs + n];
                b.x = (ka < N_) ? bx : 0.f;
                b.y = (kb < N_) ? by : 0.f;
                acc = __builtin_amdgcn_wmma_f32_16x16x4_f32(
                          false, a, false, b, (short)0, acc, false, false);
            }
            const int mbase = m0 + ((lane >> 4) << 3);
#pragma unroll
            for (int j = 0; j < 8; ++j) {
                int m = mbase + j;
                if (m < N_) Out[m * SX + n] = acc[j];
            }
        }
    }
}

// -----------------------------------------------------------------------------
// Out(19 x NO) = Hin(19 x K) @ W(K x NO) + bias [+ resid]   (K % 4 == 0)
// -----------------------------------------------------------------------------
template <int K, int NO>
__device__ __forceinline__ void matmulW(int lane, const float* __restrict__ Hin, int h_rs,
                                        const float* __restrict__ W,
                                        const float* __restrict__ bias,
                                        const float* __restrict__ resid, int res_rs,
                                        float* __restrict__ Out, int out_rs) {
    const int mloc = lane & 15;
    const int kk   = (lane >> 4) << 1;
    for (int m0 = 0; m0 < N_; m0 += 16) {
        for (int n0 = 0; n0 < NO; n0 += 16) {
            v8f acc = {0.f, 0.f, 0.f, 0.f, 0.f, 0.f, 0.f, 0.f};
            const int n = n0 + mloc;
            for (int k0 = 0; k0 < K; k0 += 4) {
                const int m   = m0 + mloc;
                const int ka  = k0 + kk;     // always < K (K multiple of 4)
                const int mcl = (m < N_) ? m : (N_ - 1);
                v2f a, b;
                float ax = Hin[mcl * h_rs + ka];
                float ay = Hin[mcl * h_rs + ka + 1];
                a.x = (m < N_) ? ax : 0.f;
                a.y = (m < N_) ? ay : 0.f;
                b.x = W[ka * NO + n];
                b.y = W[(ka + 1) * NO + n];
                acc = __builtin_amdgcn_wmma_f32_16x16x4_f32(
                          false, a, false, b, (short)0, acc, false, false);
            }
            const int mbase = m0 + ((lane >> 4) << 3);
            const float bn = bias[n];
#pragma unroll
            for (int j = 0; j < 8; ++j) {
                int m = mbase + j;
                if (m < N_) {
                    float v = acc[j] + bn;
                    if (resid) v += resid[m * res_rs + n];
                    Out[m * out_rs + n] = v;
                }
            }
        }
    }
}

// -----------------------------------------------------------------------------
// SSGC: Acc = 0.05*Xin + 0.475*(Tn^T Xin) + 0.475*(Tn^T)^2 Xin
// -----------------------------------------------------------------------------
template <int NC>
__device__ __forceinline__ void ssg_apply(int lane, const float* __restrict__ Xin, int xin_rs,
                                          float* __restrict__ Xb0, float* __restrict__ Xb1,
                                          float* __restrict__ Acc, const float* __restrict__ Tn) {
    for (int i = lane; i < N_ * NC; i += 32) {
        int r = i / NC, c = i - r * NC;
        Acc[r * SX + c] = 0.05f * Xin[r * xin_rs + c];
    }
    __syncthreads();
    matmulT<NC>(lane, Tn, Xin, xin_rs, Xb1);            // hop 1
    __syncthreads();
    for (int i = lane; i < N_ * NC; i += 32) {
        int r = i / NC, c = i - r * NC;
        Acc[r * SX + c] += 0.475f * Xb1[r * SX + c];
    }
    __syncthreads();
    matmulT<NC>(lane, Tn, Xb1, SX, Xb0);                // hop 2
    __syncthreads();
    for (int i = lane; i < N_ * NC; i += 32) {
        int r = i / NC, c = i - r * NC;
        Acc[r * SX + c] += 0.475f * Xb0[r * SX + c];
    }
    __syncthreads();
}

// -----------------------------------------------------------------------------
// One wave32 workgroup per graph.
// -----------------------------------------------------------------------------
__global__ void __launch_bounds__(32)
gcm_kernel(const float* __restrict__ features, const float* __restrict__ adjacency,
           const float* __restrict__ p_ew, const float* __restrict__ p_eb,
           const float* __restrict__ W0, const float* __restrict__ b0,
           const float* __restrict__ W1, const float* __restrict__ b1,
           const float* __restrict__ ln0_g, const float* __restrict__ ln0_b,
           const float* __restrict__ ln1_g, const float* __restrict__ ln1_b,
           float* __restrict__ out) {
    __shared__ float sAw[N_ * SA];
    __shared__ float sS [N_ * SA];     // Laplacian -> Jacobi iterate
    __shared__ float sV [N_ * SA];     // eigenvectors (columns)
    __shared__ float sTn[N_ * SA];
    __shared__ float sX0[N_ * SX];     // X / hop ping buffer
    __shared__ float sX1[N_ * SX];     // hop pong buffer
    __shared__ float sAc[N_ * SX];     // SSGC accumulator
    __shared__ float sH [N_ * SHS];    // h after first block
    __shared__ float sEv[N_];
    __shared__ float sDis[N_], sDis2[N_], sDgh[N_];
    __shared__ int   sPerm[N_];
    __shared__ int   sCol[KEIG];
    __shared__ float sSgn[KEIG];

    const int lane = threadIdx.x;
    const int g = blockIdx.x;
    const int b = g / T_;
    const int t = g - b * T_;
    const float ew = p_ew[0], eb = p_eb[0];

    const float* __restrict__ adj   = adjacency + (size_t)g * (N_ * N_);
    const float* __restrict__ featg = features + (((size_t)b * N_) * T_ + t) * D_;
    float* __restrict__ outg        = out + (((size_t)b * N_) * T_ + t) * D_;

    // Prefetch the 19 feature rows (scattered, 256B each) so they are hot in
    // cache by the time the eigensolve finishes.  -> global_prefetch_b8
#pragma unroll
    for (int n = 0; n < N_; ++n)
        __builtin_prefetch(featg + (size_t)n * T_ * D_ + lane * 2, 0, 3);

    // ---- Aw = softplus(ew*adj + eb) masked --------------------------------
    for (int i = lane; i < N_ * N_; i += 32) {
        int r = i / N_, c = i - r * N_;
        float a = adj[i];
        sAw[r * SA + c] = (a > 0.f) ? softplus_f(ew * a + eb) : 0.f;
    }
    __syncthreads();

    // ---- dis (off-diag row sums), diag_hat --------------------------------
    if (lane < N_) {
        float s = 0.f;
        for (int c = 0; c < N_; ++c)
            if (c != lane) s += sAw[lane * SA + c];
        sDis[lane] = (s > 0.f) ? (1.0f / sqrtf(s)) : 0.f;
        float dg = adj[lane * N_ + lane];
        sDgh[lane] = (dg > 0.f) ? softplus_f(ew * dg + eb) : 1.f;
    }
    __syncthreads();

    // ---- symmetrized Laplacian S = (L + L^T)/2 ; V = I --------------------
    for (int i = lane; i < N_ * N_; i += 32) {
        int r = i / N_, c = i - r * N_;
        float arc = (r == c) ? 0.f : sAw[r * SA + c];
        float acr = (r == c) ? 0.f : sAw[c * SA + r];
        float lrc = ((r == c) ? 1.f : 0.f) - sDis[r] * arc * sDis[c];
        float lcr = ((r == c) ? 1.f : 0.f) - sDis[c] * acr * sDis[r];
        sS[r * SA + c] = 0.5f * (lrc + lcr);
        sV[r * SA + c] = (r == c) ? 1.f : 0.f;
    }
    __syncthreads();

    // ---- cyclic Jacobi eigensolver (wave-lockstep, lanes = rows/cols) -----
    for (int sweep = 0; sweep < 14; ++sweep) {
        float po = 0.f;
        if (lane < N_)
            for (int r = 0; r < N_; ++r)
                if (r != lane) { float v = sS[r * SA + lane]; po += v * v; }
        for (int off = 16; off > 0; off >>= 1) po += __shfl_xor(po, off, 32);
        if (po < 1e-10f) break;                                  // uniform
        for (int p = 0; p < N_ - 1; ++p) {
            for (int q = p + 1; q < N_; ++q) {
                float apq = sS[p * SA + q];                      // uniform read
                if (fabsf(apq) > 1e-12f) {
                    float app = sS[p * SA + p], aqq = sS[q * SA + q];
                    float tau = (aqq - app) / (2.f * apq);
                    float tt  = ((tau >= 0.f) ? 1.f : -1.f) /
                                (fabsf(tau) + sqrtf(1.f + tau * tau));
                    float c = 1.0f / sqrtf(1.f + tt * tt);
                    float s = tt * c;
                    if (lane < N_) {                            // A <- A J, V <- V J
                        float akp = sS[lane * SA + p], akq = sS[lane * SA + q];
                        sS[lane * SA + p] = c * akp - s * akq;
                        sS[lane * SA + q] = s * akp + c * akq;
                        float vkp = sV[lane * SA + p], vkq = sV[lane * SA + q];
                        sV[lane * SA + p] = c * vkp - s * vkq;
                        sV[lane * SA + q] = s * vkp + c * vkq;
                    }
                    __syncthreads();
                    if (lane < N_) {                            // A <- J^T A
                        float apk = sS[p * SA + lane], aqk = sS[q * SA + lane];
                        sS[p * SA + lane] = c * apk - s * aqk;
                        sS[q * SA + lane] = s * apk + c * aqk;
                    }
                    __syncthreads();
                    if (lane == 0) { sS[p * SA + q] = 0.f; sS[q * SA + p] = 0.f; }
                    __syncthreads();
                }
            }
        }
    }

    // ---- sort eigenvalues ascending, pick columns 1..16, fix signs --------
    if (lane < N_) sEv[lane] = sS[lane * SA + lane];
    __syncthreads();
    if (lane < N_) {
        float dk = sEv[lane];
        int rank = 0;
        for (int j = 0; j < N_; ++j) {
            float dj = sEv[j];
            rank += (dj < dk) || (dj == dk && j < lane);
        }
        sPerm[rank] = lane;
    }
    __syncthreads();
    if (lane < KEIG) {
        int col = sPerm[lane + 1];
        sCol[lane] = col;
        float best = -1.f, sg = 1.f;
        for (int r = 0; r < N_; ++r) {
            float v = sV[r * SA + col];
            float av = fabsf(v);
            if (av > best) { best = av; sg = (v >= 0.f) ? 1.f : -1.f; }
        }
        sSgn[lane] = sg;
    }
    __syncthreads();

    // ---- Tn = dis2[r] * A_hat[r,c] * dis2[c]  (dis2 from column sums) -----
    if (lane < N_) {
        float s = 0.f;
        for (int r = 0; r < N_; ++r)
            s += (r == lane) ? sDgh[lane] : sAw[r * SA + lane];
        sDis2[lane] = (s > 0.f) ? (1.0f / sqrtf(s)) : 0.f;
    }
    __syncthreads();
    for (int i = lane; i < N_ * N_; i += 32) {
        int r = i / N_, c = i - r * N_;
        float ah = (r == c) ? sDgh[r] : sAw[r * SA + c];
        sTn[r * SA + c] = sDis2[r] * ah * sDis2[c];
    }

    // ---- X0 = concat(x, pe)  (19 x 80) ------------------------------------
    for (int i = lane; i < N_ * D_; i += 32) {
        int n = i >> 6, d = i & 63;
        sX0[n * SX + d] = featg[(size_t)n * T_ * D_ + d];
    }
    __syncthreads();
    for (int i = lane; i < N_ * KEIG; i += 32) {
        int n = i >> 4, k = i & 15;
        sX0[n * SX + D_ + k] = sSgn[k] * sV[n * SA + sCol[k]];
    }
    __syncthreads();

    // ---- block 0: SSGC(80) -> W0 -> LN -> GELU ----------------------------
    ssg_apply<NC0>(lane, sX0, SX, sX0, sX1, sAc, sTn);
    matmulW<NC0, D_>(lane, sAc, SX, W0, b0, nullptr, 0, sH, SHS);
    __syncthreads();
    if (lane < N_) {
        float mu = 0.f;
        for (int d = 0; d < D_; ++d) mu += sH[lane * SHS + d];
        mu *= (1.f / 64.f);
        float var = 0.f;
        for (int d = 0; d < D_; ++d) { float v = sH[lane * SHS + d] - mu; var += v * v; }
        var *= (1.f / 64.f);
        float inv = 1.0f / sqrtf(var + 1e-5f);
        for (int d = 0; d < D_; ++d) {
            float v = (sH[lane * SHS + d] - mu) * inv * ln0_g[d] + ln0_b[d];
            sH[lane * SHS + d] = gelu_f(v);
        }
    }
    __syncthreads();

    // ---- block 1: SSGC(64) -> W1 + residual -> LN -> GELU -----------------
    ssg_apply<D_>(lane, sH, SHS, sX0, sX1, sAc, sTn);
    matmulW<D_, D_>(lane, sAc, SX, W1, b1, sH, SHS, sX1, SX);
    __syncthreads();
    if (lane < N_) {
        float mu = 0.f;
        for (int d = 0; d < D_; ++d) mu += sX1[lane * SX + d];
        mu *= (1.f / 64.f);
        float var = 0.f;
        for (int d = 0; d < D_; ++d) { float v = sX1[lane * SX + d] - mu; var += v * v; }
        var *= (1.f / 64.f);
        float inv = 1.0f / sqrtf(var + 1e-5f);
        for (int d = 0; d < D_; ++d) {
            float v = (sX1[lane * SX + d] - mu) * inv * ln1_g[d] + ln1_b[d];
            sH[lane * SHS + d] = gelu_f(v);
        }
    }
    __syncthreads();

    // ---- store (B, N, T, D): same addressing pattern as the feature load --
    for (int i = lane; i < N_ * D_; i += 32) {
        int n = i >> 6, d = i & 63;
        outg[(size_t)n * T_ * D_ + d] = sH[n * SHS + d];
    }
}

extern "C" void kernel_launch(void* const* d_in, const int* in_sizes, int n_in,
                              void* d_out, int out_size, void* d_ws, size_t ws_size,
                              hipStream_t stream) {
    (void)in_sizes; (void)n_in; (void)out_size; (void)d_ws; (void)ws_size;
    gcm_kernel<<<G_, 32, 0, stream>>>(
        (const float*)d_in[0],  // features
        (const float*)d_in[1],  // adjacency
        (const float*)d_in[2],  // edge_w
        (const float*)d_in[3],  // edge_b
        (const float*)d_in[4],  // W0
        (const float*)d_in[5],  // b0
        (const float*)d_in[6],  // W1
        (const float*)d_in[7],  // b1
        (const float*)d_in[8],  // ln0_g
        (const float*)d_in[9],  // ln0_b
        (const float*)d_in[10], // ln1_g
        (const float*)d_in[11], // ln1_b
        (float*)d_out);
}